// YOLOF_5557687681626
// MI455X (gfx1250) — compile-verified
//
#include <hip/hip_runtime.h>
#include <stdint.h>

#define N_ANCH 327680
#define C_CLS  80
#define TOTAL  (N_ANCH * C_CLS)      // 26,214,400
#define TOTAL4 (TOTAL / 4)           // 6,553,600 float4s
#define KTOP   1000
#define CAP    2048
#define BINS   65536

__device__ __forceinline__ unsigned keyOf(float f) {
    unsigned u = __float_as_uint(f);
    return (u & 0x80000000u) ? ~u : (u | 0x80000000u);   // monotonic: bigger float -> bigger key
}

// ---------------- 1) zero workspace ----------------
__global__ void k_init(unsigned* ws_zero, int nwords) {
    int i = blockIdx.x * blockDim.x + threadIdx.x;
    if (i < nwords) ws_zero[i] = 0u;
}

// ---------------- 2) histogram of high 16 key bits ----------------
__global__ void k_hist_hi(const float* __restrict__ cls, unsigned* __restrict__ hist) {
    int stride = gridDim.x * blockDim.x;
    for (int i = blockIdx.x * blockDim.x + threadIdx.x; i < TOTAL4; i += stride) {
        float4 v = reinterpret_cast<const float4*>(cls)[i];
        atomicAdd(&hist[keyOf(v.x) >> 16], 1u);
        atomicAdd(&hist[keyOf(v.y) >> 16], 1u);
        atomicAdd(&hist[keyOf(v.z) >> 16], 1u);
        atomicAdd(&hist[keyOf(v.w) >> 16], 1u);
    }
}

// ---------------- 4) histogram of low 16 bits restricted to bin b1 ----------------
__global__ void k_hist_lo(const float* __restrict__ cls, const unsigned* __restrict__ sel,
                          unsigned* __restrict__ hist2) {
    unsigned b1 = sel[0];
    int stride = gridDim.x * blockDim.x;
    for (int i = blockIdx.x * blockDim.x + threadIdx.x; i < TOTAL4; i += stride) {
        float4 v = reinterpret_cast<const float4*>(cls)[i];
        unsigned k0 = keyOf(v.x), k1 = keyOf(v.y), k2 = keyOf(v.z), k3 = keyOf(v.w);
        if ((k0 >> 16) == b1) atomicAdd(&hist2[k0 & 0xFFFFu], 1u);
        if ((k1 >> 16) == b1) atomicAdd(&hist2[k1 & 0xFFFFu], 1u);
        if ((k2 >> 16) == b1) atomicAdd(&hist2[k2 & 0xFFFFu], 1u);
        if ((k3 >> 16) == b1) atomicAdd(&hist2[k3 & 0xFFFFu], 1u);
    }
}

// ---------------- 3/5) find bin containing rank-K from the top ----------------
// pass 0: rank = KTOP, writes sel[0]=bin, sel[1]=residual rank
// pass 1: rank = sel[1], writes sel[2] = full 32-bit threshold key
__global__ void k_select(const unsigned* __restrict__ hist, unsigned* __restrict__ sel, int pass) {
    __shared__ unsigned ssum[1024];
    __shared__ unsigned sabove[1024];
    int t = threadIdx.x;
    unsigned rank = (pass == 0) ? (unsigned)KTOP : sel[1];
    unsigned local = 0;
    int base = t * 64;
    for (int b = 0; b < 64; ++b) local += hist[base + b];
    ssum[t] = local;
    __syncthreads();
    if (t == 0) {
        unsigned run = 0;
        for (int tt = 1023; tt >= 0; --tt) { sabove[tt] = run; run += ssum[tt]; }
    }
    __syncthreads();
    unsigned above = sabove[t];
    if (above < rank && above + ssum[t] >= rank) {
        unsigned cnt = above;
        for (int b = base + 63; b >= base; --b) {
            unsigned h = hist[b];
            if (cnt < rank && cnt + h >= rank) {
                if (pass == 0) { sel[0] = (unsigned)b; sel[1] = rank - cnt; }
                else           { sel[2] = (sel[0] << 16) | (unsigned)b; }
                break;
            }
            cnt += h;
        }
    }
}

// ---------------- 6) compact all elements with key >= threshold ----------------
__global__ void k_collect(const float* __restrict__ cls, unsigned* __restrict__ sel,
                          unsigned* __restrict__ ckey, unsigned* __restrict__ cidx) {
    unsigned T = sel[2];
    int stride = gridDim.x * blockDim.x;
    for (int i = blockIdx.x * blockDim.x + threadIdx.x; i < TOTAL4; i += stride) {
        float4 v = reinterpret_cast<const float4*>(cls)[i];
        unsigned base = 4u * (unsigned)i;
        unsigned kk[4] = { keyOf(v.x), keyOf(v.y), keyOf(v.z), keyOf(v.w) };
        #pragma unroll
        for (int c = 0; c < 4; ++c) {
            if (kk[c] >= T) {
                unsigned p = atomicAdd(&sel[3], 1u);
                if (p < CAP) { ckey[p] = kk[c]; cidx[p] = base + c; }
            }
        }
    }
}

// ---------------- 7) bitonic sort (desc key, asc idx) + gather ----------------
__global__ void __launch_bounds__(1024)
k_sort_gather(const float* __restrict__ cls, const float* __restrict__ box,
              const unsigned* __restrict__ sel, const unsigned* __restrict__ ckey,
              const unsigned* __restrict__ cidx,
              float* __restrict__ o_score, int* __restrict__ o_label,
              int* __restrict__ o_valid, float* __restrict__ o_box) {
    __shared__ unsigned long long s[CAP];
    int t = threadIdx.x;
    unsigned M = sel[3]; if (M > (unsigned)CAP) M = CAP;
    for (int e = t; e < CAP; e += 1024) {
        unsigned long long c = 0ull;
        if (e < (int)M)
            c = (((unsigned long long)ckey[e]) << 32) |
                (unsigned long long)(0xFFFFFFFFu - cidx[e]);   // desc key, asc idx
        s[e] = c;
    }
    __syncthreads();
    for (int k = 2; k <= CAP; k <<= 1) {
        for (int j = k >> 1; j > 0; j >>= 1) {
            int low = t & (j - 1);
            int i = ((t & ~(j - 1)) << 1) | low;
            int p = i | j;
            bool dir = ((i & k) == 0);            // descending blocks
            unsigned long long a = s[i], b = s[p];
            bool sw = dir ? (a < b) : (a > b);
            if (sw) { s[i] = b; s[p] = a; }
            __syncthreads();
        }
    }
    // extraction: threads >= KTOP write deterministic zeros (NMS kernel reads them)
    unsigned long long c = s[t];
    if (t < KTOP && c != 0ull) {
        unsigned idx = 0xFFFFFFFFu - (unsigned)(c & 0xFFFFFFFFull);
        float logit = cls[idx];
        float score = 1.0f / (1.0f + expf(-logit));
        int label  = (int)(idx % C_CLS);
        int anchor = (int)(idx / C_CLS);
        float4 b = reinterpret_cast<const float4*>(box)[anchor];
        o_score[t] = score;
        o_label[t] = label;
        o_valid[t] = (score > 0.05f) ? 1 : 0;
        reinterpret_cast<float4*>(o_box)[t] = b;
    } else {
        o_score[t] = 0.0f; o_label[t] = 0; o_valid[t] = 0;
        float4 z; z.x = z.y = z.z = z.w = 0.0f;
        reinterpret_cast<float4*>(o_box)[t] = z;
    }
}

// ---------------- 8) NMS + outputs (async global->LDS staging, CDNA5 path) ----------------
__global__ void __launch_bounds__(1024)
k_nms_out(const float* __restrict__ o_score, const int* __restrict__ o_label,
          const int* __restrict__ o_valid, const float* __restrict__ o_box,
          const int* __restrict__ img_h, const int* __restrict__ img_w,
          float* __restrict__ out) {
    __shared__ float s_box[4096];   // 1024 boxes * 4 (original, un-offset)
    __shared__ float s_lab[1024];
    __shared__ int   s_keep[1024];
    int t = threadIdx.x;

    // force the LDS array's address to escape so the "memory" clobber pins it
    float* sb = s_box;
    asm volatile("" : "+v"(sb) :: "memory");

    // CDNA5 async copy engine: 16 KB of boxes, one b128 per lane, ASYNCcnt-tracked
    unsigned lds_addr = (unsigned)(uintptr_t)sb + (unsigned)t * 16u;
    unsigned voff = (unsigned)t * 16u;
    asm volatile("global_load_async_to_lds_b128 %0, %1, %2"
                 :: "v"(lds_addr), "v"(voff), "s"(o_box) : "memory");

    s_lab[t]  = (float)o_label[t];
    s_keep[t] = o_valid[t];

    asm volatile("s_wait_asynccnt 0" ::: "memory");
    __syncthreads();

    // per-thread class-offset box (labels * (IMG+1) trick for class-aware NMS)
    float off = s_lab[t] * 1025.0f;
    float x1 = s_box[4 * t + 0] + off, y1 = s_box[4 * t + 1] + off;
    float x2 = s_box[4 * t + 2] + off, y2 = s_box[4 * t + 3] + off;
    float area = (x2 - x1) * (y2 - y1);
    bool my = (s_keep[t] != 0);

    for (int i = 0; i < KTOP; ++i) {
        __syncthreads();                      // publish previous round's suppressions
        if (s_keep[i]) {
            float oi = s_lab[i] * 1025.0f;
            float bx1 = s_box[4 * i + 0] + oi, by1 = s_box[4 * i + 1] + oi;
            float bx2 = s_box[4 * i + 2] + oi, by2 = s_box[4 * i + 3] + oi;
            float ai = (bx2 - bx1) * (by2 - by1);
            float xx1 = fmaxf(bx1, x1), yy1 = fmaxf(by1, y1);
            float xx2 = fminf(bx2, x2), yy2 = fminf(by2, y2);
            float inter = fmaxf(xx2 - xx1, 0.0f) * fmaxf(yy2 - yy1, 0.0f);
            float iou = inter / (ai + area - inter + 1e-9f);
            if (t > i && my && iou > 0.6f) { my = false; s_keep[t] = 0; }
        }
    }
    __syncthreads();

    if (t < KTOP) {
        float w = (float)img_w[0], h = (float)img_h[0];
        float b0 = s_box[4 * t + 0] / w, b1 = s_box[4 * t + 1] / h;
        float b2 = s_box[4 * t + 2] / w, b3 = s_box[4 * t + 3] / h;
        out[4 * t + 0] = fminf(fmaxf(b0, 0.0f), 1.0f);
        out[4 * t + 1] = fminf(fmaxf(b1, 0.0f), 1.0f);
        out[4 * t + 2] = fminf(fmaxf(b2, 0.0f), 1.0f);
        out[4 * t + 3] = fminf(fmaxf(b3, 0.0f), 1.0f);
        out[4 * KTOP + t] = my ? o_score[t] : 0.0f;   // scores_out
        out[5 * KTOP + t] = s_lab[t];                 // labels (as float)
        out[6 * KTOP + t] = my ? 1.0f : 0.0f;         // keep
    }
}

extern "C" void kernel_launch(void* const* d_in, const int* in_sizes, int n_in,
                              void* d_out, int out_size, void* d_ws, size_t ws_size,
                              hipStream_t stream) {
    const float* cls = (const float*)d_in[0];   // [1, N, C] f32
    const float* box = (const float*)d_in[1];   // [1, N, 4] f32
    const int* img_h = (const int*)d_in[2];
    const int* img_w = (const int*)d_in[3];
    float* out = (float*)d_out;

    // workspace layout (words)
    unsigned* hist1 = (unsigned*)d_ws;              // 65536
    unsigned* hist2 = hist1 + BINS;                 // 65536
    unsigned* sel   = hist2 + BINS;                 // 16  ([0]=b1 [1]=rank2 [2]=T [3]=counter)
    unsigned* ckey  = sel + 16;                     // 2048
    unsigned* cidx  = ckey + CAP;                   // 2048
    float*    o_score = (float*)(cidx + CAP);       // 1024
    int*      o_label = (int*)(o_score + 1024);     // 1024
    int*      o_valid = o_label + 1024;             // 1024
    float*    o_box   = (float*)(o_valid + 1024);   // 4096 (16B-aligned offset)

    int zwords = 2 * BINS + 16;
    k_init<<<(zwords + 255) / 256, 256, 0, stream>>>((unsigned*)d_ws, zwords);
    k_hist_hi<<<2048, 256, 0, stream>>>(cls, hist1);
    k_select<<<1, 1024, 0, stream>>>(hist1, sel, 0);
    k_hist_lo<<<2048, 256, 0, stream>>>(cls, sel, hist2);
    k_select<<<1, 1024, 0, stream>>>(hist2, sel, 1);
    k_collect<<<2048, 256, 0, stream>>>(cls, sel, ckey, cidx);
    k_sort_gather<<<1, 1024, 0, stream>>>(cls, box, sel, ckey, cidx,
                                          o_score, o_label, o_valid, o_box);
    k_nms_out<<<1, 1024, 0, stream>>>(o_score, o_label, o_valid, o_box,
                                      img_h, img_w, out);
    (void)in_sizes; (void)n_in; (void)out_size; (void)ws_size;
}